// SAGENet_69045894250552
// MI455X (gfx1250) — compile-verified
//
#include <hip/hip_runtime.h>
#include <math.h>

typedef float v2f __attribute__((ext_vector_type(2)));
typedef float v8f __attribute__((ext_vector_type(8)));

#define D 96
#define NGRAPH 64

// ---------------- utility kernels ----------------

__global__ void zero_f32(float* __restrict__ p, int n) {
    int i = blockIdx.x * blockDim.x + threadIdx.x;
    if (i < n) p[i] = 0.0f;
}

__global__ void count_deg(const int* __restrict__ dst, float* __restrict__ cnt, int E) {
    int e = blockIdx.x * blockDim.x + threadIdx.x;
    if (e < E) atomicAdd(&cnt[dst[e]], 1.0f);
}

__global__ void make_inv(float* __restrict__ cnt, int n) {
    int i = blockIdx.x * blockDim.x + threadIdx.x;
    if (i < n) cnt[i] = 1.0f / fmaxf(cnt[i], 1.0f);
}

// grid = (ceil(E/256), 24): x -> edge, y -> 4-feature chunk. Pure float4 gather
// + 4 f32 atomics, all L2-resident (x/s arrays are ~19 MB each vs 192 MB L2).
__global__ void scatter_add(const float* __restrict__ xin, const int* __restrict__ src,
                            const int* __restrict__ dst, float* __restrict__ s, int E) {
    const int e = blockIdx.x * blockDim.x + threadIdx.x;
    if (e >= E) return;
    const int f4 = blockIdx.y * 4;
    const int sr = src[e];
    const int ds = dst[e];
    const float4 v = *(const float4*)(xin + (long long)sr * D + f4);
    float* o = s + (long long)ds * D + f4;
    atomicAdd(o + 0, v.x);
    atomicAdd(o + 1, v.y);
    atomicAdd(o + 2, v.z);
    atomicAdd(o + 3, v.w);
}

// ---------------- fused SAGE layer: h = ELU( (s*inv) @ Wl^T + x @ Wr^T + bl ) ----------------
// One wave computes a 16-row x 96-col tile with V_WMMA_F32_16X16X4_F32.
// Fragment layouts (ISA 7.12.2, 32-bit):
//   A (16x4): lanes 0-15 -> M=0..15, K={k0,k0+1}; lanes 16-31 -> K={k0+2,k0+3}.
//   B (4x16): lanes 0-15 -> N=0..15, K={k0,k0+1}; lanes 16-31 -> K={k0+2,k0+3}.
//   C/D (16x16): vgpr v -> M=v (lanes 0-15) / M=v+8 (lanes 16-31), N=lane%16.
// Two-deep software pipeline: fragment loads for step k+1 are issued before the
// 12 WMMAs of step k, so each WMMA batch only waits on loads a full step old.
__global__ __launch_bounds__(256)
void sage_gemm_elu(const float* __restrict__ s, const float* __restrict__ inv,
                   const float* __restrict__ x,
                   const float* __restrict__ Wl, const float* __restrict__ Wr,
                   const float* __restrict__ bl, float* __restrict__ h,
                   int ntiles, int nnodes) {
    const int wave = threadIdx.x >> 5;
    const int tile = blockIdx.x * 8 + wave;
    if (tile >= ntiles) return;            // wave-uniform: EXEC stays all-1s for WMMA
    const int lane  = threadIdx.x & 31;
    const int half2 = (lane >> 4) * 2;     // K sub-pair select
    const int l16   = lane & 15;

    const int row0 = tile * 16;
    int mrow = row0 + l16;
    if (mrow >= nnodes) mrow = nnodes - 1; // clamp (no-op when N % 16 == 0)

    const float scale = inv[mrow];
    // pre-biased lane pointers: all inner-loop loads use pure immediate offsets
    const float* __restrict__ srow = s + (long long)mrow * D + half2;
    const float* __restrict__ xrow = x + (long long)mrow * D + half2;
    const float* __restrict__ wlp  = Wl + l16 * D + half2;   // + n*16*D + k0
    const float* __restrict__ wrp  = Wr + l16 * D + half2;

    v8f acc[6] = {};
    v2f a1f[2], a2f[2], bfl[2][6], bfr[2][6];

    // prologue: stage k-step 0
    {
        v2f t = *(const v2f*)(srow);
        a1f[0] = t * scale;                 // fold mean-divide into A
        a2f[0] = *(const v2f*)(xrow);
#pragma unroll
        for (int n = 0; n < 6; ++n) {
            bfl[0][n] = *(const v2f*)(wlp + n * 16 * D);
            bfr[0][n] = *(const v2f*)(wrp + n * 16 * D);
        }
    }

#pragma unroll
    for (int kk = 0; kk < D / 4; ++kk) {
        const int cur = kk & 1, nxt = cur ^ 1;
        if (kk + 1 < D / 4) {               // prefetch next k-step fragments
            const int k0 = (kk + 1) * 4;
            v2f t = *(const v2f*)(srow + k0);
            a1f[nxt] = t * scale;
            a2f[nxt] = *(const v2f*)(xrow + k0);
#pragma unroll
            for (int n = 0; n < 6; ++n) {
                bfl[nxt][n] = *(const v2f*)(wlp + n * 16 * D + k0);
                bfr[nxt][n] = *(const v2f*)(wrp + n * 16 * D + k0);
            }
        }
#pragma unroll
        for (int n = 0; n < 6; ++n) {
            acc[n] = __builtin_amdgcn_wmma_f32_16x16x4_f32(
                false, a1f[cur], false, bfl[cur][n], (short)0, acc[n], false, false);
            acc[n] = __builtin_amdgcn_wmma_f32_16x16x4_f32(
                false, a2f[cur], false, bfr[cur][n], (short)0, acc[n], false, false);
        }
    }

#pragma unroll
    for (int n = 0; n < 6; ++n) {
        const int ncol = n * 16 + l16;
        const float bias = bl[ncol];
#pragma unroll
        for (int v = 0; v < 8; ++v) {
            const int m = row0 + v + half2 * 4;       // half*8
            if (m < nnodes) {
                float val = acc[n][v] + bias;
                val = (val > 0.0f) ? val : (__expf(val) - 1.0f);   // ELU(alpha=1)
                h[(long long)m * D + ncol] = val;
            }
        }
    }
}

// ---------------- segment max pooling (monotonic unsigned key trick) ----------------

__device__ __forceinline__ unsigned fkey(float f) {
    unsigned b = __float_as_uint(f);
    return (b & 0x80000000u) ? ~b : (b | 0x80000000u);
}
__device__ __forceinline__ float funkey(unsigned k) {
    unsigned b = (k & 0x80000000u) ? (k & 0x7FFFFFFFu) : ~k;
    return __uint_as_float(b);
}

__global__ void pool_max(const float* __restrict__ h, const int* __restrict__ batch,
                         unsigned* __restrict__ pooled, int nnodes) {
    int t = blockIdx.x * blockDim.x + threadIdx.x;
    if (t >= nnodes * D) return;
    int i = t / D, f = t % D;
    int g = batch[i];
    atomicMax(&pooled[g * D + f], fkey(h[(long long)i * D + f]));
}

// ---------------- MLP head: relu(relu(pooled@Wm1^T+bm1)@Wm2^T+bm2) ----------------

__global__ __launch_bounds__(256)
void mlp_head(const unsigned* __restrict__ pooled_u,
              const float* __restrict__ Wm1, const float* __restrict__ bm1,
              const float* __restrict__ Wm2, const float* __restrict__ bm2,
              float* __restrict__ out) {
    __shared__ float pbuf[NGRAPH * D];
    __shared__ float hbuf[NGRAPH * D];
    const int tid = threadIdx.x;
    for (int i = tid; i < NGRAPH * D; i += 256) {
        float p = funkey(pooled_u[i]);
        if (!(fabsf(p) < 3.0e38f)) p = 0.0f;   // kill NaN/Inf (empty segments)
        pbuf[i] = p;
    }
    __syncthreads();
    for (int i = tid; i < NGRAPH * D; i += 256) {
        int g = i / D, n = i % D;
        float acc = bm1[n];
        for (int k = 0; k < D; ++k) acc += pbuf[g * D + k] * Wm1[n * D + k];
        hbuf[i] = fmaxf(acc, 0.0f);
    }
    __syncthreads();
    for (int i = tid; i < NGRAPH * 8; i += 256) {
        int g = i / 8, n = i % 8;
        float acc = bm2[n];
        for (int k = 0; k < D; ++k) acc += hbuf[g * D + k] * Wm2[n * D + k];
        out[i] = fmaxf(acc, 0.0f);
    }
}

// ---------------- launch ----------------

extern "C" void kernel_launch(void* const* d_in, const int* in_sizes, int n_in,
                              void* d_out, int out_size, void* d_ws, size_t ws_size,
                              hipStream_t stream) {
    const float* x     = (const float*)d_in[0];
    const int*   eidx  = (const int*)d_in[1];
    const int*   batch = (const int*)d_in[2];
    const float* Wl1   = (const float*)d_in[3];
    const float* bl1   = (const float*)d_in[4];
    const float* Wr1   = (const float*)d_in[5];
    const float* Wl2   = (const float*)d_in[6];
    const float* bl2   = (const float*)d_in[7];
    const float* Wr2   = (const float*)d_in[8];
    const float* Wm1   = (const float*)d_in[9];
    const float* bm1   = (const float*)d_in[10];
    const float* Wm2   = (const float*)d_in[11];
    const float* bm2   = (const float*)d_in[12];

    const int N = in_sizes[0] / D;        // 50000
    const int E = in_sizes[1] / 2;        // 800000
    const int* src = eidx;
    const int* dst = eidx + E;

    const int NF = N * D;
    const int ntiles = (N + 15) / 16;

    // workspace layout (floats)
    float* ws   = (float*)d_ws;
    float* cnt  = ws;                      // N    (becomes inv-degree)
    float* s    = ws + ((N + 63) & ~63);   // N*D  (scatter sums)
    float* h1   = s + NF;                  // N*D
    float* h2   = h1 + NF;                 // N*D
    unsigned* pooled = (unsigned*)(h2 + NF); // 64*D keys

    const int B = 256;
    dim3 blk(B);
    auto g1 = [&](int n) { return dim3((n + B - 1) / B); };
    const dim3 gsc((E + B - 1) / B, D / 4);         // scatter: edge x feature-chunk
    const dim3 ggemm((ntiles + 7) / 8);

    // degree + inverse
    zero_f32<<<g1(N), blk, 0, stream>>>(cnt, N);
    count_deg<<<g1(E), blk, 0, stream>>>(dst, cnt, E);
    make_inv<<<g1(N), blk, 0, stream>>>(cnt, N);

    // layer 1
    zero_f32<<<g1(NF), blk, 0, stream>>>(s, NF);
    scatter_add<<<gsc, blk, 0, stream>>>(x, src, dst, s, E);
    sage_gemm_elu<<<ggemm, blk, 0, stream>>>(s, cnt, x, Wl1, Wr1, bl1, h1, ntiles, N);

    // layer 2
    zero_f32<<<g1(NF), blk, 0, stream>>>(s, NF);
    scatter_add<<<gsc, blk, 0, stream>>>(h1, src, dst, s, E);
    sage_gemm_elu<<<ggemm, blk, 0, stream>>>(s, cnt, h1, Wl2, Wr2, bl2, h2, ntiles, N);

    // pooling (key 0 decodes to NaN -> zeroed in head, matches isfinite guard)
    zero_f32<<<g1(NGRAPH * D), blk, 0, stream>>>((float*)pooled, NGRAPH * D);
    pool_max<<<g1(NF), blk, 0, stream>>>(h2, batch, pooled, N);

    // head
    mlp_head<<<dim3(1), blk, 0, stream>>>(pooled, Wm1, bm1, Wm2, bm2, (float*)d_out);
}